// SpectralBias_76098230550921
// MI455X (gfx1250) — compile-verified
//
#include <hip/hip_runtime.h>
#include <math.h>

// ---------------- constants matching the reference ----------------
#define KFREQ   6
#define MCOMP   2
#define LSEQ    2048
#define NHEAD   16
#define DHEAD   64
#define HIDDEN  128
#define NOUT    11          // 5*M + 1
#define BETA    0.5f
#define CSCALE  0.01f
#define WMIN    32.0f
#define WMAX    256.0f
#define DSTARMX 2047.0f
#define LR      0.003125f   // RAMP_LAMBDA / TAU = 0.2/64
#define INVTAU  0.015625f   // 1/64

typedef __attribute__((ext_vector_type(2))) float v2f;
typedef __attribute__((ext_vector_type(8))) float v8f;

struct Omegas { float w[KFREQ]; };

__device__ __forceinline__ float sigmoidf_(float x) { return 1.0f / (1.0f + expf(-x)); }
__device__ __forceinline__ float softplusf_(float x) {
    return fmaxf(x, 0.0f) + log1pf(expf(-fabsf(x)));
}

// ---------------------------------------------------------------------------
// Kernel A: build the kv-side cos/sin table, interleaved for WMMA B-matrix.
// btab[(n*2 + o)*8 + 2*j + t] = Bk(4j + 2o + t, n)
// Bk(k,n) = k<6: cos(w_k n); k<12: sin(w_{k-6} n); k==12: 1; else 0
// ---------------------------------------------------------------------------
__global__ void sb_btab_kernel(float* __restrict__ btab, Omegas om) {
    int n = blockIdx.x * blockDim.x + threadIdx.x;
    if (n >= LSEQ) return;
    float nf = (float)n;
    float cv[KFREQ], sv[KFREQ];
#pragma unroll
    for (int k = 0; k < KFREQ; ++k) {
        float a = om.w[k] * nf;
        cv[k] = cosf(a);
        sv[k] = sinf(a);
    }
#pragma unroll
    for (int kidx = 0; kidx < 16; ++kidx) {
        float v;
        if (kidx < 6)       v = cv[kidx];
        else if (kidx < 12) v = sv[kidx - 6];
        else if (kidx == 12) v = 1.0f;
        else                v = 0.0f;
        int t = kidx & 1, o = (kidx >> 1) & 1, j = kidx >> 2;
        btab[(size_t)(n * 2 + o) * 8 + 2 * j + t] = v;
    }
}

// ---------------------------------------------------------------------------
// Kernel B: per-query MLP -> spectral coefficients (A-matrix, interleaved)
// atab[(q*2 + o)*8 + 2*j + t] = Acoef(4j+2o+t)  for global query index q=h*L+l
// Acoef[0..5]=beta*Ccoef, [6..11]=beta*Scoef, [12]=-beta*b0, [13..15]=0
// sl[q] = softplus(feats[10]) / TAU
// ---------------------------------------------------------------------------
__global__ void __launch_bounds__(256)
sb_mlp_kernel(const float* __restrict__ q,
              const float* __restrict__ W1, const float* __restrict__ b1,
              const float* __restrict__ W2, const float* __restrict__ b2,
              float* __restrict__ atab, float* __restrict__ sl, Omegas om) {
    int t = blockIdx.x * blockDim.x + threadIdx.x;
    if (t >= NHEAD * LSEQ) return;
    int l = t & (LSEQ - 1);

    const float* qv = q + (size_t)t * DHEAD;
    float qr[DHEAD];
#pragma unroll
    for (int i = 0; i < DHEAD; ++i) qr[i] = qv[i];

    float feats[NOUT];
#pragma unroll
    for (int o = 0; o < NOUT; ++o) feats[o] = b2[o];

    for (int e = 0; e < HIDDEN; ++e) {
        const float* w1r = W1 + e * DHEAD;   // uniform -> scalar loads
        float d = b1[e];
#pragma unroll
        for (int i = 0; i < DHEAD; ++i) d = fmaf(w1r[i], qr[i], d);
        float he = d * sigmoidf_(d);         // silu
#pragma unroll
        for (int o = 0; o < NOUT; ++o) feats[o] = fmaf(W2[o * HIDDEN + e], he, feats[o]);
    }

    float cc[MCOMP], pi[MCOMP], ds[MCOMP], wd[MCOMP], ph[MCOMP];
    pi[0] = sigmoidf_(feats[1] - feats[6]);  // softmax over 2
    pi[1] = 1.0f - pi[0];
#pragma unroll
    for (int m = 0; m < MCOMP; ++m) {
        cc[m] = softplusf_(feats[5 * m + 0]) * CSCALE;
        ds[m] = DSTARMX * sigmoidf_(feats[5 * m + 2]);
        wd[m] = WMIN + (WMAX - WMIN) * sigmoidf_(feats[5 * m + 3]);
        ph[m] = feats[5 * m + 4];
    }
    sl[t] = softplusf_(feats[10]) * INVTAU;

    float A[16];
    float lf = (float)l;
    float b0 = 0.0f;
#pragma unroll
    for (int k = 0; k < KFREQ; ++k) {
        float wk = om.w[k];
        float Ck = 0.0f, Sk = 0.0f;
#pragma unroll
        for (int m = 0; m < MCOMP; ++m) {
            float wkw = wk * wd[m];
            float env = expf(-0.5f * wkw * wkw);
            float amp = pi[m] * cc[m] * env;
            float arg = (lf - ds[m]) * wk + ph[m];
            Ck = fmaf(amp, cosf(arg), Ck);
            Sk = fmaf(amp, sinf(arg), Sk);
            b0 = fmaf(amp, cosf(ds[m] * wk - ph[m]), b0);
        }
        A[k]     = BETA * Ck;
        A[6 + k] = BETA * Sk;
    }
    A[12] = -BETA * b0;
    A[13] = 0.0f; A[14] = 0.0f; A[15] = 0.0f;

    size_t base = (size_t)t * 16;
#pragma unroll
    for (int kidx = 0; kidx < 16; ++kidx) {
        int tt = kidx & 1, o = (kidx >> 1) & 1, j = kidx >> 2;
        atab[base + (size_t)o * 8 + 2 * j + tt] = A[kidx];
    }
}

// ---------------------------------------------------------------------------
// Kernel C: [H,L,L] bias via V_WMMA_F32_16X16X4_F32.
// Block = 256 threads = 8 waves. Each wave -> one 16(q) x 16(kv) tile.
// Block covers 16 q x 128 kv. Grid = H * (L/16) * (L/128) = 32768 blocks.
// Branchless epilogue; slope preloaded as 2x b128; stores use const offsets.
// ---------------------------------------------------------------------------
__global__ void __launch_bounds__(256)
sb_bias_kernel(const float* __restrict__ atab, const float* __restrict__ sl,
               const float* __restrict__ btab, float* __restrict__ out) {
    int bid  = blockIdx.x;
    int hd   = bid >> 11;            // / (128*16)
    int rem  = bid & 2047;
    int qt   = rem >> 4;
    int kvb  = rem & 15;
    int wave = threadIdx.x >> 5;
    int lane = threadIdx.x & 31;

    int q0   = qt << 4;
    int kv0  = (kvb << 7) + (wave << 4);
    int nloc = lane & 15;
    int o    = lane >> 4;            // K-half selector (A/B), row-half (C/D)
    int n    = kv0 + nloc;

    // Base output pointer for this lane: row q0+8o, column n; row stride LSEQ
    // (8192-byte constant offsets fit the 24-bit signed immediate).
    float* op = out + ((size_t)(hd * LSEQ + q0 + (o << 3))) * LSEQ + n;

    // Fully-masked tile (wave-uniform): write zeros, skip WMMA.
    if (q0 + 15 < kv0) {
#pragma unroll
        for (int r = 0; r < 8; ++r) op[(size_t)r * LSEQ] = 0.0f;
        return;
    }

    // A: row M = lane&15 (query q0+M), half o -> K pairs {4j+2o, 4j+2o+1}
    const float4* ap4 = (const float4*)(atab + ((size_t)(hd * LSEQ + q0 + nloc) * 16) + o * 8);
    float4 alo = ap4[0], ahi = ap4[1];
    // B: col N = n, same K interleave
    const float4* bp4 = (const float4*)(btab + ((size_t)n * 2 + o) * 8);
    float4 blo = bp4[0], bhi = bp4[1];
    // Slopes for this lane's 8 rows are contiguous: sl[hd*L + q0 + 8o .. +7]
    const float4* sp4 = (const float4*)(sl + (size_t)hd * LSEQ + q0 + (o << 3));
    float4 slo = sp4[0], shi = sp4[1];
    float sv[8] = {slo.x, slo.y, slo.z, slo.w, shi.x, shi.y, shi.z, shi.w};

    v2f a0 = {alo.x, alo.y}, a1 = {alo.z, alo.w}, a2 = {ahi.x, ahi.y}, a3 = {ahi.z, ahi.w};
    v2f b0 = {blo.x, blo.y}, b1 = {blo.z, blo.w}, b2 = {bhi.x, bhi.y}, b3 = {bhi.z, bhi.w};

    v8f acc = {};
    acc = __builtin_amdgcn_wmma_f32_16x16x4_f32(false, a0, false, b0, (short)0, acc, false, false);
    acc = __builtin_amdgcn_wmma_f32_16x16x4_f32(false, a1, false, b1, (short)0, acc, false, false);
    acc = __builtin_amdgcn_wmma_f32_16x16x4_f32(false, a2, false, b2, (short)0, acc, false, false);
    acc = __builtin_amdgcn_wmma_f32_16x16x4_f32(false, a3, false, b3, (short)0, acc, false, false);
    // acc now holds beta*(fourier - b0) for rows q0 + r + 8*o, col n.

    float dbase = (float)(q0 + (o << 3) - n);   // delta at r = 0
#pragma unroll
    for (int r = 0; r < 8; ++r) {
        float delta = dbase + (float)r;
        float ramp  = 1.0f - __expf(-LR * delta);     // computed unconditionally
        float val   = fmaf(ramp, acc[r], -sv[r] * delta);
        val = (delta >= 0.0f) ? val : 0.0f;           // branchless causal mask
        op[(size_t)r * LSEQ] = val;
    }
}

// ---------------------------------------------------------------------------
extern "C" void kernel_launch(void* const* d_in, const int* in_sizes, int n_in,
                              void* d_out, int out_size, void* d_ws, size_t ws_size,
                              hipStream_t stream) {
    const float* q  = (const float*)d_in[0];
    const float* W1 = (const float*)d_in[1];
    const float* b1 = (const float*)d_in[2];
    const float* W2 = (const float*)d_in[3];
    const float* b2 = (const float*)d_in[4];
    float* out = (float*)d_out;

    // workspace layout
    float* atab = (float*)d_ws;                     // 16*2048*16 = 524288 floats
    float* slw  = atab + (size_t)NHEAD * LSEQ * 16; // 32768 floats
    float* btab = slw + (size_t)NHEAD * LSEQ;       // 32768 floats

    // omegas: logspace(log10(2pi/1e6), log10(2pi/2048), 6) in double
    Omegas om;
    const double PI2 = 6.283185307179586476925286766559;
    double la = log10(PI2 / 1000000.0);
    double lb = log10(PI2 / 2048.0);
    for (int k = 0; k < KFREQ; ++k)
        om.w[k] = (float)pow(10.0, la + (lb - la) * ((double)k / 5.0));

    sb_btab_kernel<<<(LSEQ + 255) / 256, 256, 0, stream>>>(btab, om);
    sb_mlp_kernel<<<(NHEAD * LSEQ) / 256, 256, 0, stream>>>(q, W1, b1, W2, b2, atab, slw, om);
    sb_bias_kernel<<<NHEAD * (LSEQ / 16) * (LSEQ / 128), 256, 0, stream>>>(atab, slw, btab, out);
}